// GHMC_1829656068729
// MI455X (gfx1250) — compile-verified
//
#include <hip/hip_runtime.h>

#define GHM_BINS 10

typedef float v2f __attribute__((ext_vector_type(2)));
typedef float v8f __attribute__((ext_vector_type(8)));

// ---------------------------------------------------------------------------
// Pass 1: streaming kernel. Each thread owns a private LDS histogram column;
// count and bce-sum are fused in one float2 slot -> single ds_load_b64 +
// ds_store_b64 per element (conflict-free: 8B stride covers all 64 banks).
// Deterministic (no atomics, fixed per-thread order). Emits per-block
// partials: d_ws[block*20 + 0..9] = counts, [10..19] = sum(valid*bce).
// ---------------------------------------------------------------------------
__global__ __launch_bounds__(256) void ghmc_partial_kernel(
    const float* __restrict__ pred,
    const float* __restrict__ targ,
    const float* __restrict__ lw,
    float* __restrict__ partial,
    long long ne)   // total elements
{
    __shared__ float2 hist[GHM_BINS][256];   // .x = count, .y = sum(valid*bce)

    const int tid = threadIdx.x;
    #pragma unroll
    for (int b = 0; b < GHM_BINS; ++b)
        hist[b][tid] = make_float2(0.0f, 0.0f);
    __syncthreads();

    const long long ne4    = ne >> 2;                 // float4 groups
    const long long stride = (long long)gridDim.x * blockDim.x;
    long long i = (long long)blockIdx.x * blockDim.x + tid;

    const float4* p4 = (const float4*)pred;
    const float4* t4 = (const float4*)targ;
    const float4* w4 = (const float4*)lw;

    for (; i < ne4; i += stride) {
        float4 pv = p4[i];
        float4 tv = t4[i];
        float4 wv = w4[i];
        float pe[4] = {pv.x, pv.y, pv.z, pv.w};
        float te[4] = {tv.x, tv.y, tv.z, tv.w};
        float we[4] = {wv.x, wv.y, wv.z, wv.w};
        #pragma unroll
        for (int j = 0; j < 4; ++j) {
            float p = pe[j], t = te[j];
            float v  = (we[j] > 0.0f) ? 1.0f : 0.0f;
            float ap = fabsf(p);
            float e  = __expf(-ap);                    // exp(-|p|) in (0,1]
            float s1 = 1.0f + e;
            float inv = __builtin_amdgcn_rcpf(s1);     // v_rcp_f32
            float sig = (p >= 0.0f) ? inv : e * inv;   // sigmoid(p)
            float g   = fabsf(sig - t);
            int b = (int)(g * 10.0f);                  // g >= 0: trunc==floor
            b = (b > GHM_BINS - 1) ? (GHM_BINS - 1) : (b < 0 ? 0 : b);
            // stable BCE-with-logits: max(p,0) - p*t + log(1+exp(-|p|))
            float bce = fmaxf(p, 0.0f) - p * t + __logf(s1);
            float2 hv = hist[b][tid];                  // ds_load_b64
            hv.x += v;
            hv.y = fmaf(v, bce, hv.y);
            hist[b][tid] = hv;                         // ds_store_b64
        }
    }

    // scalar tail (ne % 4), handled by block 0 only
    long long tail_base = ne4 << 2;
    if (blockIdx.x == 0 && tid < (int)(ne - tail_base)) {
        long long k = tail_base + tid;
        float p = pred[k], t = targ[k];
        float v  = (lw[k] > 0.0f) ? 1.0f : 0.0f;
        float ap = fabsf(p);
        float e  = __expf(-ap);
        float s1 = 1.0f + e;
        float inv = __builtin_amdgcn_rcpf(s1);
        float sig = (p >= 0.0f) ? inv : e * inv;
        float g   = fabsf(sig - t);
        int b = (int)(g * 10.0f);
        b = (b > GHM_BINS - 1) ? (GHM_BINS - 1) : (b < 0 ? 0 : b);
        float bce = fmaxf(p, 0.0f) - p * t + __logf(s1);
        float2 hv = hist[b][tid];
        hv.x += v;
        hv.y = fmaf(v, bce, hv.y);
        hist[b][tid] = hv;
    }

    __syncthreads();

    // reduce 256 columns -> 1 value per (bin, kind); fixed serial order
    if (tid < 2 * GHM_BINS) {
        int b = (tid < GHM_BINS) ? tid : (tid - GHM_BINS);
        float acc = 0.0f;
        if (tid < GHM_BINS) {
            for (int c = 0; c < 256; ++c) acc += hist[b][c].x;
        } else {
            for (int c = 0; c < 256; ++c) acc += hist[b][c].y;
        }
        partial[(long long)blockIdx.x * (2 * GHM_BINS) + tid] = acc;
    }
}

// ---------------------------------------------------------------------------
// Pass 2: 320 threads = 16 rows x 20 columns. Stage 1: strided partial sums
// over block records. Stage 2: fixed 16-way fold. Stage 3: wave 0 only
// (EXEC all-ones) computes Sum_b w[b]*S[b] with 4 chained
// V_WMMA_F32_16X16X4_F32 (B = all-ones, accumulate in C; D[0][0] on lane 0).
// ---------------------------------------------------------------------------
__global__ __launch_bounds__(320) void ghmc_final_kernel(
    const float* __restrict__ partial, int nblocks, float* __restrict__ out)
{
    __shared__ float part[16][2 * GHM_BINS];
    __shared__ float red[2 * GHM_BINS];

    const int t   = threadIdx.x;          // 0..319
    const int col = t % (2 * GHM_BINS);   // 0..19
    const int row = t / (2 * GHM_BINS);   // 0..15

    float acc = 0.0f;
    for (int g = row; g < nblocks; g += 16)
        acc += partial[(long long)g * (2 * GHM_BINS) + col];
    part[row][col] = acc;
    __syncthreads();

    if (t < 2 * GHM_BINS) {
        float a = 0.0f;
        #pragma unroll
        for (int rr = 0; rr < 16; ++rr) a += part[rr][t];
        red[t] = a;
    }
    __syncthreads();

    if (t < 32) {                          // wave 0 only: EXEC all ones here
        const int lane = t;
        float cnt[GHM_BINS], S[GHM_BINS];
        #pragma unroll
        for (int b = 0; b < GHM_BINS; ++b) {
            cnt[b] = red[b];
            S[b]   = red[GHM_BINS + b];
        }

        float tot = 0.0f, n = 0.0f;
        #pragma unroll
        for (int b = 0; b < GHM_BINS; ++b) tot += cnt[b];
        tot = fmaxf(tot, 1.0f);
        #pragma unroll
        for (int b = 0; b < GHM_BINS; ++b) n += (cnt[b] > 0.0f) ? 1.0f : 0.0f;

        // t16[i] = w_per_bin[i] * S[i], padded to 16 with zeros
        float t16[16];
        #pragma unroll
        for (int b = 0; b < GHM_BINS; ++b)
            t16[b] = (cnt[b] > 0.0f) ? (tot / fmaxf(cnt[b], 1.0f)) * S[b] : 0.0f;
        #pragma unroll
        for (int b = GHM_BINS; b < 16; ++b) t16[b] = 0.0f;

        // A 16x4 f32 layout: lanes 0-15 hold K=0 (vgpr0), K=1 (vgpr1);
        // lanes 16-31 hold K=2, K=3. Broadcast across M rows, B = ones
        // => D[m][n] = Sum_k A[*][k] (layout-insensitive in B).
        const bool hi = (lane >= 16);
        v2f bmat; bmat.x = 1.0f; bmat.y = 1.0f;
        v8f c = {0.0f, 0.0f, 0.0f, 0.0f, 0.0f, 0.0f, 0.0f, 0.0f};
        #pragma unroll
        for (int ch = 0; ch < 4; ++ch) {
            v2f a;
            a.x = hi ? t16[4 * ch + 2] : t16[4 * ch + 0];
            a.y = hi ? t16[4 * ch + 3] : t16[4 * ch + 1];
            c = __builtin_amdgcn_wmma_f32_16x16x4_f32(
                    false, a, false, bmat, (short)0, c, false, false);
        }

        if (lane == 0) {
            // loss = (Sum_b w[b]*S[b]) / (max(n,1) * tot) ; LOSS_WEIGHT = 1
            out[0] = c[0] / (fmaxf(n, 1.0f) * tot);
        }
    }
}

extern "C" void kernel_launch(void* const* d_in, const int* in_sizes, int n_in,
                              void* d_out, int out_size, void* d_ws, size_t ws_size,
                              hipStream_t stream) {
    const float* pred = (const float*)d_in[0];
    const float* targ = (const float*)d_in[1];
    const float* lw   = (const float*)d_in[2];
    long long ne = (long long)in_sizes[0];

    // one partial record = 20 floats per block; clamp grid to scratch capacity
    int max_blocks = (int)(ws_size / (size_t)(2 * GHM_BINS * sizeof(float)));
    int nblocks = 1024;
    if (nblocks > max_blocks) nblocks = max_blocks;
    if (nblocks < 1) nblocks = 1;

    ghmc_partial_kernel<<<nblocks, 256, 0, stream>>>(pred, targ, lw,
                                                     (float*)d_ws, ne);
    ghmc_final_kernel<<<1, 320, 0, stream>>>((const float*)d_ws, nblocks,
                                             (float*)d_out);
}